// SageConvolution_47837345743433
// MI455X (gfx1250) — compile-verified
//
#include <hip/hip_runtime.h>

// SAGE graph convolution for gfx1250 (MI455X).
//   h   = x @ W_l                      (workspace, fp32)
//   out = x @ W_r + bias_l + bias_r    (init)
//   out[dst] += w[e] * h[src]          (edge scatter, L2-resident f32 atomics)

typedef __attribute__((ext_vector_type(2))) float v2f;
typedef __attribute__((ext_vector_type(8))) float v8f;

#define DIM 128

// ---------------------------------------------------------------------------
// Kernel A: fused dual GEMM using V_WMMA_F32_16X16X4_F32.
// Block = 256 threads = 8 waves. Each block: 16 node rows x 128 output cols.
// Wave w owns the 16-column tile [16*w, 16*w+16). K loop: 128 in steps of 4.
// ---------------------------------------------------------------------------
__global__ __launch_bounds__(256) void sage_gemm_dual(
    const float* __restrict__ x,    // [N, 128]
    const float* __restrict__ Wl,   // [128, 128]
    const float* __restrict__ Wr,   // [128, 128]
    const float* __restrict__ bl,   // [128]
    const float* __restrict__ br,   // [128]
    float* __restrict__ h,          // [N, 128] workspace (x @ Wl)
    float* __restrict__ out,        // [N, 128]
    int nNodes)
{
    const int lane    = threadIdx.x & 31;
    const int wave    = threadIdx.x >> 5;       // 0..7 -> column tile
    const int rowBase = blockIdx.x * 16;
    const int colBase = wave * 16;

    const int m     = lane & 15;                // A row / B,C,D column index
    const int khalf = (lane >> 4) * 2;          // lanes 0-15: K+{0,1}; 16-31: K+{2,3}
    const int col   = colBase + m;

    // Clamp the load row so EXEC stays all-1s through the WMMA sequence
    // (N_NODES % 16 == 0 here, but keep it robust).
    int arow = rowBase + m;
    if (arow >= nNodes) arow = nNodes - 1;
    const float* xrow = x + (size_t)arow * DIM;

    v8f accL = {};  // x @ Wl tile
    v8f accR = {};  // x @ Wr tile

#pragma unroll 4
    for (int kk = 0; kk < DIM; kk += 4) {
        const int krow = kk + khalf;

        // A fragment: 16x4 f32 (ISA 7.12.2: lane&15 = M, halves carry K pairs)
        v2f a;
        a.x = xrow[krow + 0];
        a.y = xrow[krow + 1];

        // B fragments: 4x16 f32, N striped across lanes, K split across halves
        v2f bL, bR;
        bL.x = Wl[(size_t)(krow + 0) * DIM + col];
        bL.y = Wl[(size_t)(krow + 1) * DIM + col];
        bR.x = Wr[(size_t)(krow + 0) * DIM + col];
        bR.y = Wr[(size_t)(krow + 1) * DIM + col];

        // (neg_a, A, neg_b, B, c_mod, C, reuse_a, reuse_b)
        accL = __builtin_amdgcn_wmma_f32_16x16x4_f32(
            false, a, false, bL, (short)0, accL, false, false);
        accR = __builtin_amdgcn_wmma_f32_16x16x4_f32(
            false, a, false, bR, (short)0, accR, false, false);
    }

    // out init carries both biases; h carries none (bias_l is added once, here).
    const float biasSum = bl[col] + br[col];
    const int   mHalf   = (lane >> 4) * 8;      // C/D: VGPR i -> M=i or M=i+8
    const size_t base   = (size_t)(rowBase + mHalf) * DIM + col;

    if (rowBase + 16 <= nNodes) {
        // Hot path: uniform full tile -> straight-line stores, immediate offsets.
        float* hp = h   + base;
        float* op = out + base;
#pragma unroll
        for (int i = 0; i < 8; ++i) {
            hp[(size_t)i * DIM] = accL[i];
            op[(size_t)i * DIM] = accR[i] + biasSum;
        }
    } else {
        // Cold tail path (unused for N % 16 == 0, kept for robustness).
#pragma unroll
        for (int i = 0; i < 8; ++i) {
            const int r = rowBase + i + mHalf;
            if (r < nNodes) {
                const size_t idx = (size_t)r * DIM + col;
                h[idx]   = accL[i];
                out[idx] = accR[i] + biasSum;
            }
        }
    }
}

// ---------------------------------------------------------------------------
// Kernel B: edge scatter. One wave per edge; lane i owns floats [4i, 4i+4).
// 512 B coalesced gather of h[src] (L2-resident: h = 51 MB < 192 MB L2),
// then 4 global_atomic_add_f32 per lane into out[dst].
// ---------------------------------------------------------------------------
__global__ __launch_bounds__(256) void sage_edge_scatter(
    const float* __restrict__ h,     // [N, 128]
    const int*   __restrict__ src,   // [E]
    const int*   __restrict__ dst,   // [E]
    const float* __restrict__ ew,    // [E]
    float* __restrict__ out,         // [N, 128]
    int nEdges)
{
    const int wid  = (int)((blockIdx.x * (unsigned)blockDim.x + threadIdx.x) >> 5);
    const int lane = threadIdx.x & 31;
    if (wid >= nEdges) return;

    const int   s  = src[wid];
    const int   d  = dst[wid];
    const float wt = ew[wid];

    const float4 v = *((const float4*)(h + (size_t)s * DIM) + lane);

    float* op = out + (size_t)d * DIM + lane * 4;
    atomicAdd(op + 0, v.x * wt);
    atomicAdd(op + 1, v.y * wt);
    atomicAdd(op + 2, v.z * wt);
    atomicAdd(op + 3, v.w * wt);
}

// ---------------------------------------------------------------------------
extern "C" void kernel_launch(void* const* d_in, const int* in_sizes, int n_in,
                              void* d_out, int out_size, void* d_ws, size_t ws_size,
                              hipStream_t stream) {
    const float* x   = (const float*)d_in[0];
    const int*   es  = (const int*)  d_in[1];
    const int*   ed  = (const int*)  d_in[2];
    const float* ewt = (const float*)d_in[3];
    const float* Wl  = (const float*)d_in[4];
    const float* bl  = (const float*)d_in[5];
    const float* Wr  = (const float*)d_in[6];
    const float* br  = (const float*)d_in[7];

    float* out = (float*)d_out;
    float* h   = (float*)d_ws;              // [nNodes, 128] fp32 scratch

    const int nNodes = in_sizes[0] / DIM;
    const int nEdges = in_sizes[1];

    // Phase 1: dual GEMM + bias init (overwrites all of d_out every call).
    const int rowTiles = (nNodes + 15) / 16;
    sage_gemm_dual<<<rowTiles, 256, 0, stream>>>(x, Wl, Wr, bl, br, h, out, nNodes);

    // Phase 2: edge aggregation (8 edges per 256-thread block).
    const int edgeBlocks = (nEdges + 7) / 8;
    sage_edge_scatter<<<edgeBlocks, 256, 0, stream>>>(h, es, ed, ewt, out, nEdges);
}